// Part_2048_83519934038063
// MI455X (gfx1250) — compile-verified
//
#include <hip/hip_runtime.h>
#include <math.h>

// Problem constants (B=1, N=12, C=3, H=W=1024)
constexpr int W_DIM = 1024;
constexpr int H_DIM = 1024;
constexpr int C_DIM = 3;
constexpr int N_DIM = 12;
constexpr int HW_BITS = 10;                 // log2(W_DIM)
constexpr int PLANE = H_DIM * W_DIM;        // 1<<20
constexpr long long XS_ELEMS = (long long)N_DIM * C_DIM * PLANE;
constexpr long long OS_ELEMS = (long long)N_DIM * PLANE;

typedef float v4f __attribute__((ext_vector_type(4)));

// Main kernel: one thread per (transform m, 4 consecutive output pixels).
// Block = 256 threads = one full output row; grid = (H, N).
// Bilinear sample of img (zero padding) for 3 channels + analytic ones-mask.
// Outputs leave as non-temporal b128 stores (201 MB streamed; keeps the 12 MB
// input image resident in L2 for the gathers).
__global__ __launch_bounds__(256) void warp_affine_kernel(
    const float* __restrict__ img,    // [3, H, W]
    const float* __restrict__ theta,  // [N, 2, 3] row-major
    float* __restrict__ xs,           // [N, 3, H, W]
    float* __restrict__ os)           // [N, H, W]
{
    const int m  = blockIdx.y;                   // wave-uniform transform id
    const int h  = blockIdx.x;                   // wave-uniform row
    const int w0 = (int)threadIdx.x << 2;        // first of 4 pixels

    // theta load is uniform per block -> scalar cache
    const float t00 = theta[m * 6 + 0];
    const float t01 = theta[m * 6 + 1];
    const float t02 = theta[m * 6 + 2];
    const float t10 = theta[m * 6 + 3];
    const float t11 = theta[m * 6 + 4];
    const float t12 = theta[m * 6 + 5];

    // Row-invariant parts of grid = theta * [gx, gy, 1]
    const float gy = ((float)h + 0.5f) * (2.0f / (float)H_DIM) - 1.0f;
    const float by = t01 * gy + t02;   // grid_x = t00*gx + by
    const float cy = t11 * gy + t12;   // grid_y = t10*gx + cy

    const float* p0 = img;
    const float* p1 = img + PLANE;
    const float* p2 = img + 2 * PLANE;

    v4f v0 = {0.0f, 0.0f, 0.0f, 0.0f};
    v4f v1 = {0.0f, 0.0f, 0.0f, 0.0f};
    v4f v2 = {0.0f, 0.0f, 0.0f, 0.0f};
    v4f vm = {0.0f, 0.0f, 0.0f, 0.0f};

#pragma unroll
    for (int i = 0; i < 4; ++i) {
        const int w = w0 + i;
        const float gx = ((float)w + 0.5f) * (2.0f / (float)W_DIM) - 1.0f;

        const float grid_x = t00 * gx + by;
        const float grid_y = t10 * gx + cy;

        // Input-space sample position
        const float ix = ((grid_x + 1.0f) * (float)W_DIM - 1.0f) * 0.5f;
        const float iy = ((grid_y + 1.0f) * (float)H_DIM - 1.0f) * 0.5f;

        const float x0f = floorf(ix);
        const float y0f = floorf(iy);
        const float x1f = x0f + 1.0f;
        const float y1f = y0f + 1.0f;

        const float wx1 = ix - x0f, wx0 = 1.0f - wx1;
        const float wy1 = iy - y0f, wy0 = 1.0f - wy1;

        const bool vx0 = (x0f >= 0.0f) && (x0f < (float)W_DIM);
        const bool vx1 = (x1f >= 0.0f) && (x1f < (float)W_DIM);
        const bool vy0 = (y0f >= 0.0f) && (y0f < (float)H_DIM);
        const bool vy1 = (y1f >= 0.0f) && (y1f < (float)H_DIM);

        // Masked bilinear weights (zero padding outside)
        const float m00 = (vx0 && vy0) ? (wx0 * wy0) : 0.0f;
        const float m10 = (vx1 && vy0) ? (wx1 * wy0) : 0.0f;
        const float m01 = (vx0 && vy1) ? (wx0 * wy1) : 0.0f;
        const float m11 = (vx1 && vy1) ? (wx1 * wy1) : 0.0f;

        // ones-image sample = sum of masked weights (no gather needed)
        const float mask = m00 + m10 + m01 + m11;
        vm[i] = mask;

        if (mask > 0.0f) {
            // Clamped integer indices (coords bounded, safe int conversion)
            int xi0 = (int)x0f; xi0 = xi0 < 0 ? 0 : (xi0 > W_DIM - 1 ? W_DIM - 1 : xi0);
            int xi1 = (int)x1f; xi1 = xi1 < 0 ? 0 : (xi1 > W_DIM - 1 ? W_DIM - 1 : xi1);
            int yi0 = (int)y0f; yi0 = yi0 < 0 ? 0 : (yi0 > H_DIM - 1 ? H_DIM - 1 : yi0);
            int yi1 = (int)y1f; yi1 = yi1 < 0 ? 0 : (yi1 > H_DIM - 1 ? H_DIM - 1 : yi1);

            const int r0  = (yi0 << HW_BITS);
            const int r1  = (yi1 << HW_BITS);
            const int i00 = r0 + xi0, i10 = r0 + xi1;
            const int i01 = r1 + xi0, i11 = r1 + xi1;

            v0[i] = m00 * p0[i00] + m10 * p0[i10] + m01 * p0[i01] + m11 * p0[i11];
            v1[i] = m00 * p1[i00] + m10 * p1[i10] + m01 * p1[i01] + m11 * p1[i11];
            v2[i] = m00 * p2[i00] + m10 * p2[i10] + m01 * p2[i01] + m11 * p2[i11];
        }
    }

    // Streaming non-temporal b128 stores (16B aligned: w0 % 4 == 0)
    const long long row = ((long long)h << HW_BITS) + w0;
    float* xs_m = xs + (long long)m * C_DIM * PLANE + row;
    __builtin_nontemporal_store(v0, (v4f*)(xs_m));
    __builtin_nontemporal_store(v1, (v4f*)(xs_m + PLANE));
    __builtin_nontemporal_store(v2, (v4f*)(xs_m + 2 * PLANE));
    __builtin_nontemporal_store(vm, (v4f*)(os + (long long)m * PLANE + row));
}

// Tail kernel: copy z and compute closed-form inverse affine.
__global__ __launch_bounds__(32) void affine_tail_kernel(
    const float* __restrict__ theta,  // [N, 2, 3]
    float* __restrict__ z_out,        // [N, 2, 3]
    float* __restrict__ inv_out)      // [N, 2, 3]
{
    const int m = threadIdx.x;
    if (m >= N_DIM) return;

    const float a  = theta[m * 6 + 0];
    const float b  = theta[m * 6 + 1];
    const float tx = theta[m * 6 + 2];
    const float c  = theta[m * 6 + 3];
    const float d  = theta[m * 6 + 4];
    const float ty = theta[m * 6 + 5];

    z_out[m * 6 + 0] = a;  z_out[m * 6 + 1] = b;  z_out[m * 6 + 2] = tx;
    z_out[m * 6 + 3] = c;  z_out[m * 6 + 4] = d;  z_out[m * 6 + 5] = ty;

    const float rdet = 1.0f / (a * d - b * c);
    const float ia =  d * rdet, ib = -b * rdet;
    const float ic = -c * rdet, id =  a * rdet;
    const float itx = -(ia * tx + ib * ty);
    const float ity = -(ic * tx + id * ty);

    inv_out[m * 6 + 0] = ia;  inv_out[m * 6 + 1] = ib;  inv_out[m * 6 + 2] = itx;
    inv_out[m * 6 + 3] = ic;  inv_out[m * 6 + 4] = id;  inv_out[m * 6 + 5] = ity;
}

extern "C" void kernel_launch(void* const* d_in, const int* in_sizes, int n_in,
                              void* d_out, int out_size, void* d_ws, size_t ws_size,
                              hipStream_t stream) {
    const float* x     = (const float*)d_in[0];  // (1,3,1024,1024) f32
    const float* theta = (const float*)d_in[1];  // (1,12,2,3)      f32

    float* out = (float*)d_out;
    float* xs  = out;                            // 12*3*2^20
    float* os  = out + XS_ELEMS;                 // 12*2^20
    float* z   = out + XS_ELEMS + OS_ELEMS;      // 12*6
    float* iz  = z + N_DIM * 6;                  // 12*6

    dim3 grid(H_DIM, N_DIM, 1);                  // one block per output row
    warp_affine_kernel<<<grid, dim3(256, 1, 1), 0, stream>>>(x, theta, xs, os);
    affine_tail_kernel<<<dim3(1, 1, 1), dim3(32, 1, 1), 0, stream>>>(theta, z, iz);
}